// SINDyLibrary_85779086835853
// MI455X (gfx1250) — compile-verified
//
#include <hip/hip_runtime.h>

// SINDy polynomial library, N_DIM = 32.
// Columns: [1 | z(32) | pairs i<=j (528) | triples a<=b<=c (5984) | sin(z)(32)] = 6577
#define NDIM  32
#define NPAIR 528
#define NTRIP 5984
#define NCOL  (1 + NDIM + NPAIR + NTRIP + NDIM)   // 6577
#define ROWS_PER_BLOCK 8
#define PSTRIDE 33                                 // pad to avoid LDS bank conflicts

typedef float v2f __attribute__((ext_vector_type(2)));
typedef float v8f __attribute__((ext_vector_type(8)));

struct PairTab { unsigned short v[NPAIR]; };
struct TripTab { unsigned int   v[NTRIP]; };

static constexpr PairTab make_pairs() {
    PairTab t{};
    int n = 0;
    for (int i = 0; i < NDIM; ++i)
        for (int j = i; j < NDIM; ++j)
            t.v[n++] = (unsigned short)((i << 8) | j);
    return t;
}
static constexpr TripTab make_trips() {
    TripTab t{};
    int n = 0;
    for (int a = 0; a < NDIM; ++a)
        for (int b = a; b < NDIM; ++b)
            for (int c = b; c < NDIM; ++c)
                t.v[n++] = (unsigned int)(a | (b << 8) | (c << 16));
    return t;
}

__device__ __constant__ PairTab c_pair = make_pairs();
__device__ __constant__ TripTab c_trip = make_trips();

__global__ __launch_bounds__(256)
void sindy_library_kernel(const float* __restrict__ z,
                          float* __restrict__ out, int B) {
    __shared__ float zs[ROWS_PER_BLOCK][NDIM];
    __shared__ float P [ROWS_PER_BLOCK][NDIM * PSTRIDE];

    const int lane = threadIdx.x & 31;
    const int w    = threadIdx.x >> 5;                    // wave id in block (wave32)
    const int row  = blockIdx.x * ROWS_PER_BLOCK + w;     // one row per wave
    const bool rowOK = (row < B);                         // wave-uniform -> EXEC all-1s inside

    if (rowOK) zs[w][lane] = z[(size_t)row * NDIM + lane];
    __syncthreads();

    // ---- P = z * z^T via 4x v_wmma_f32_16x16x4_f32 (rank-1: only K=0 nonzero) ----
    if (rowOK) {
        const int hi = lane >> 4;       // 0: lanes 0-15, 1: lanes 16-31
        const int lN = lane & 15;
        #pragma unroll
        for (int mi = 0; mi < 2; ++mi) {
            // A 16x4 f32: VGPR0 = K=0 (lanes 0-15), K=2 (lanes 16-31); VGPR1 = K=1/K=3.
            // Only K=0 carries data; all other K slices are zero.
            float a0 = (hi == 0) ? zs[w][mi * 16 + lN] : 0.0f;
            v2f A = {a0, 0.0f};
            #pragma unroll
            for (int ni = 0; ni < 2; ++ni) {
                float b0 = (hi == 0) ? zs[w][ni * 16 + lN] : 0.0f;
                v2f Bm = {b0, 0.0f};
                v8f acc = {};
                acc = __builtin_amdgcn_wmma_f32_16x16x4_f32(
                          /*neg_a=*/false, A, /*neg_b=*/false, Bm,
                          /*c_mod=*/(short)0, acc,
                          /*reuse_a=*/false, /*reuse_b=*/false);
                // D layout: VGPR v -> row M = v + 8*hi (within tile), N = lane&15
                const int m0 = mi * 16 + hi * 8;
                const int n  = ni * 16 + lN;
                #pragma unroll
                for (int v = 0; v < 8; ++v)
                    P[w][(m0 + v) * PSTRIDE + n] = acc[v];
            }
        }
    }
    __syncthreads();

    if (!rowOK) return;

    // ---- stream the 6577-column row: coalesced per-wave non-temporal stores ----
    float* const orow = out + (size_t)row * NCOL;
    for (int col = lane; col < NCOL; col += 32) {
        float val;
        if (col == 0) {
            val = 1.0f;
        } else if (col < 1 + NDIM) {
            val = zs[w][col - 1];
        } else if (col < 1 + NDIM + NPAIR) {
            const unsigned ij = c_pair.v[col - (1 + NDIM)];
            val = P[w][(ij >> 8) * PSTRIDE + (ij & 255u)];
        } else if (col < 1 + NDIM + NPAIR + NTRIP) {
            const unsigned t = c_trip.v[col - (1 + NDIM + NPAIR)];
            val = zs[w][t & 255u] * P[w][((t >> 8) & 255u) * PSTRIDE + (t >> 16)];
        } else {
            val = sinf(zs[w][col - (NCOL - NDIM)]);
        }
        __builtin_nontemporal_store(val, orow + col);   // output > L2: don't thrash
    }
}

extern "C" void kernel_launch(void* const* d_in, const int* in_sizes, int n_in,
                              void* d_out, int out_size, void* d_ws, size_t ws_size,
                              hipStream_t stream) {
    (void)n_in; (void)out_size; (void)d_ws; (void)ws_size;
    const float* z  = (const float*)d_in[0];
    float*       out = (float*)d_out;
    const int B = in_sizes[0] / NDIM;                 // 8192
    const int grid = (B + ROWS_PER_BLOCK - 1) / ROWS_PER_BLOCK;
    sindy_library_kernel<<<grid, 256, 0, stream>>>(z, out, B);
}